// S2Deconvolution_49770081026365
// MI455X (gfx1250) — compile-verified
//
#include <hip/hip_runtime.h>
#include <math.h>

// ---------------- problem constants ----------------
#define NLM   256           // B_OUT^2
#define NP    5456          // sum_{l<16} (2l+1)^2
#define SCALING_F (1.0f / sqrtf(98304.0f))   // 1/sqrt(24*64*16^4/32^2)

typedef __attribute__((ext_vector_type(2))) float v2f;
typedef __attribute__((ext_vector_type(8))) float v8f;

static __device__ __forceinline__ v8f wmma4(v2f a, v2f b, v8f c) {
  // V_WMMA_F32_16X16X4_F32 : D = A(16x4) * B(4x16) + C(16x16), all f32
  return __builtin_amdgcn_wmma_f32_16x16x4_f32(false, a, false, b, (short)0, c,
                                               false, false);
}

static __device__ __forceinline__ int off3(int l) { return l * (4 * l * l - 1) / 3; }

// A-fragment: A row-major [M x K], tile rows rowBase..+15, k-cols k0..k0+3
static __device__ __forceinline__ v2f ldA(const float* A, int lda, int rowBase,
                                          int k0, int lane) {
  int r  = rowBase + (lane & 15);
  int kk = k0 + ((lane >> 4) << 1);            // lanes 16-31 hold K=2,3
  v2f a; a.x = A[r * lda + kk]; a.y = A[r * lda + kk + 1]; return a;
}
// B-fragment: B row-major [K x N], tile cols colBase..+15, k-rows k0..k0+3
static __device__ __forceinline__ v2f ldB(const float* B, int ldb, int colBase,
                                          int k0, int lane) {
  int c  = colBase + (lane & 15);
  int kk = k0 + ((lane >> 4) << 1);
  v2f b; b.x = B[kk * ldb + c]; b.y = B[(kk + 1) * ldb + c]; return b;
}

// ---------------- Wigner small-d (double, device) ----------------
static __device__ double wig_d(int l, int mp, int mm, double beta) {
  double cb = cos(beta * 0.5), sb = sin(beta * 0.5);
  double pref = 0.5 * (lgamma((double)(l + mp + 1)) + lgamma((double)(l - mp + 1)) +
                       lgamma((double)(l + mm + 1)) + lgamma((double)(l - mm + 1)));
  int smin = mm - mp > 0 ? mm - mp : 0;
  int smax = (l + mm) < (l - mp) ? (l + mm) : (l - mp);
  double tot = 0.0;
  for (int s = smin; s <= smax; ++s) {
    double ln = pref - (lgamma((double)(l + mm - s + 1)) + lgamma((double)(s + 1)) +
                        lgamma((double)(mp - mm + s + 1)) + lgamma((double)(l - mp - s + 1)));
    double sgn = ((mp - mm + s) & 1) ? -1.0 : 1.0;
    tot += sgn * exp(ln) * pow(cb, (double)(2 * l + mm - mp - 2 * s)) *
           pow(sb, (double)(mp - mm + 2 * s));
  }
  return tot;
}

// ---------------- K0: build all constant tables ----------------
__global__ void k_tables(float* __restrict__ W, float* __restrict__ Dt,
                         float* __restrict__ Yre, float* __restrict__ Yim,
                         float* __restrict__ twR, float* __restrict__ twI,
                         float* __restrict__ Ec, float* __restrict__ Es) {
  const double PI = 3.14159265358979323846;
  int gid = blockIdx.x * blockDim.x + threadIdx.x;
  // --- segment 0: W_S2 [64 betas][256 lm], incl quadrature weight ---
  if (gid < 16384) {
    int b = gid >> 8, lm = gid & 255;
    int l = 0; while ((l + 1) * (l + 1) <= lm) ++l;
    int mi = lm - l * l;
    double beta = PI * (2 * b + 1) / 128.0;
    double s = 0.0;
    for (int k = 0; k < 32; ++k)
      s += sin((2.0 * b + 1.0) * (2.0 * k + 1.0) * PI / 128.0) / (2.0 * k + 1.0);
    double w = (2.0 / 32.0) * sin(PI * (2 * b + 1) / 128.0) * s * (2.0 * PI / 4096.0);
    W[gid] = (float)(w * wig_d(l, mi - l, 0, beta));
    return;
  }
  gid -= 16384;
  // --- segment 1: D_SO3 packed [32 j][5456 P], incl (2l+1) factor ---
  if (gid < 32 * NP) {
    int j = gid / NP, P = gid % NP;
    int l = 0; while (l < 15 && off3(l + 1) <= P) ++l;
    int q = P - off3(l), LL = 2 * l + 1;
    int mi = q / LL, ni = q % LL;
    double beta = PI * (2 * j + 1) / 64.0;
    Dt[gid] = (float)((double)LL * wig_d(l, mi - l, ni - l, beta));
    return;
  }
  gid -= 32 * NP;
  // --- segment 2: Y_GRID [24 g][256 lm] complex ---
  if (gid < 24 * 256) {
    int g = gid >> 8, lm = gid & 255;
    int l = 0; while ((l + 1) * (l + 1) <= lm) ++l;
    int mi = lm - l * l, m = mi - l;
    double beta  = (double)(g / 8 + 1) * PI / 24.0;
    double alpha = (double)(g % 8) * PI / 4.0;
    double val = sqrt((2.0 * l + 1.0) / (4.0 * PI)) * wig_d(l, m, 0, beta);
    Yre[gid] = (float)(val * cos(m * alpha));
    Yim[gid] = (float)(val * sin(m * alpha));
    return;
  }
  gid -= 24 * 256;
  // --- segment 3: alpha-DFT twiddles [64 a][31 u], e^{-2pi i m a/64} ---
  if (gid < 64 * 31) {
    int a = gid / 31, u = gid % 31, m = u - 15;
    double ang = 2.0 * PI * m * a / 64.0;
    twR[gid] = (float)cos(ang);
    twI[gid] = (float)(-sin(ang));
    return;
  }
  gid -= 64 * 31;
  // --- segment 4: 32x32 inverse-Fourier table rows m=-15..15 (+zero pad) ---
  if (gid < 1024) {
    int r = gid >> 5, p = gid & 31;
    if (r == 31) { Ec[gid] = 0.f; Es[gid] = 0.f; return; }
    double ang = 2.0 * PI * (double)(r - 15) * p / 32.0;
    Ec[gid] = (float)cos(ang);
    Es[gid] = (float)sin(ang);
  }
}

// ---------------- K1: DFT over alpha (only 31 bins needed) ----------------
__global__ void k_dft(const float* __restrict__ x, const float* __restrict__ twR,
                      const float* __restrict__ twI, float* __restrict__ xfR,
                      float* __restrict__ xfI) {
  __shared__ float sR[64 * 31], sI[64 * 31];
  for (int i = threadIdx.x; i < 64 * 31; i += blockDim.x) { sR[i] = twR[i]; sI[i] = twI[i]; }
  __syncthreads();
  int gid = blockIdx.x * blockDim.x + threadIdx.x;
  if (gid >= 131072 * 31) return;
  int row = gid / 31, u = gid % 31;
  const float* xr = x + row * 64;
  float ar = 0.f, ai = 0.f;
  #pragma unroll
  for (int a = 0; a < 64; ++a) {
    float xv = xr[a];
    ar = fmaf(xv, sR[a * 31 + u], ar);
    ai = fmaf(xv, sI[a * 31 + u], ai);
  }
  xfR[gid] = ar; xfI[gid] = ai;
}

// ---------------- K2: beta projection  ->  Fx_l [32L x 64] ----------------
__global__ void k_project(const float* __restrict__ W, const float* __restrict__ xfR,
                          const float* __restrict__ xfI, float* __restrict__ FxR,
                          float* __restrict__ FxI) {
  int g = blockIdx.x * blockDim.x + threadIdx.x;
  if (g >= 524288) return;
  int l = 0; while (l < 15 && 2048 * (l + 1) * (l + 1) <= g) ++l;
  int rem = g - 2048 * l * l;
  int r = rem >> 6, f = rem & 63;      // r = mi*32 + batch
  int mi = r >> 5, zb = r & 31;
  int u = mi - l + 15;
  int wcol = l * l + mi;
  int base = ((zb * 64 + f) * 64) * 31 + u;
  float ar = 0.f, ai = 0.f;
  for (int b = 0; b < 64; ++b) {
    float wv = W[b * 256 + wcol];
    ar = fmaf(wv, xfR[base + b * 31], ar);
    ai = fmaf(wv, xfI[base + b * 31], ai);
  }
  FxR[g] = ar; FxI[g] = ai;
}

// ---------------- K3: kernel SHT  ->  Fy_l [64 x 32L] ----------------
__global__ void k_y(const float* __restrict__ kern, const float* __restrict__ Yre,
                    const float* __restrict__ Yim, float* __restrict__ FyR,
                    float* __restrict__ FyI) {
  int g = blockIdx.x * blockDim.x + threadIdx.x;
  if (g >= 524288) return;
  int l = 0; while (l < 15 && 2048 * (l + 1) * (l + 1) <= g) ++l;
  int rem = g - 2048 * l * l;
  int M = 32 * (2 * l + 1);
  int f = rem / M, c = rem % M;        // c = mi*32 + f_out
  int mi = c >> 5, oo = c & 31;
  int lm = l * l + mi;
  const float* kp = kern + (f * 32 + oo) * 24;
  float ar = 0.f, ai = 0.f;
  #pragma unroll
  for (int q = 0; q < 24; ++q) {
    float kv = kp[q] * SCALING_F;
    ar = fmaf(kv, Yre[q * 256 + lm], ar);
    ai = fmaf(kv, Yim[q * 256 + lm], ai);
  }
  FyR[g] = ar; FyI[g] = ai;
}

// ---------------- K4: den = Re(Fy^H Fy)  (WMMA f32 GEMM) ----------------
__global__ void __launch_bounds__(32) k_den(const float* __restrict__ FyR,
                                            const float* __restrict__ FyI,
                                            float* __restrict__ den) {
  int l = blockIdx.z;
  int M = 32 * (2 * l + 1);
  int rb = blockIdx.x * 16, cb = blockIdx.y * 16;
  if (rb >= M || cb >= M) return;
  int lane = threadIdx.x;
  const float* BR = FyR + 2048 * l * l;
  const float* BI = FyI + 2048 * l * l;
  v8f acc = {};
  for (int k0 = 0; k0 < 64; k0 += 4) {
    v2f aR = ldB(BR, M, rb, k0, lane);     // A = Fy^T : A[r][k] = Fy[k][r]
    v2f aI = ldB(BI, M, rb, k0, lane);
    v2f bR = ldB(BR, M, cb, k0, lane);
    v2f bI = ldB(BI, M, cb, k0, lane);
    acc = wmma4(aR, bR, acc);
    acc = wmma4(aI, bI, acc);
  }
  float* dst = den + 1024 * off3(l);
  int c  = cb + (lane & 15);
  int r0 = rb + ((lane >> 4) << 3);
  #pragma unroll
  for (int v = 0; v < 8; ++v) dst[(r0 + v) * M + c] = acc[v];
}

// ---- K5: num = Fx@Fy (complex WMMA), Fz = num/den, pack (l,m,n)[z,o] ----
__global__ void __launch_bounds__(32) k_num(const float* __restrict__ FxR,
                                            const float* __restrict__ FxI,
                                            const float* __restrict__ FyR,
                                            const float* __restrict__ FyI,
                                            const float* __restrict__ den,
                                            float* __restrict__ FzR,
                                            float* __restrict__ FzI) {
  int l = blockIdx.z;
  int LL = 2 * l + 1;
  int M = 32 * LL;
  int rb = blockIdx.x * 16, cb = blockIdx.y * 16;
  if (rb >= M || cb >= M) return;
  int lane = threadIdx.x;
  const float* AR = FxR + 2048 * l * l;
  const float* AI = FxI + 2048 * l * l;
  const float* BR = FyR + 2048 * l * l;
  const float* BI = FyI + 2048 * l * l;
  v8f cRR = {}, cII = {}, cRI = {}, cIR = {};
  for (int k0 = 0; k0 < 64; k0 += 4) {
    v2f aR = ldA(AR, 64, rb, k0, lane);
    v2f aI = ldA(AI, 64, rb, k0, lane);
    v2f bR = ldB(BR, M, cb, k0, lane);
    v2f bI = ldB(BI, M, cb, k0, lane);
    cRR = wmma4(aR, bR, cRR);
    cII = wmma4(aI, bI, cII);
    cRI = wmma4(aR, bI, cRI);
    cIR = wmma4(aI, bR, cIR);
  }
  const float* dl = den + 1024 * off3(l);
  int c  = cb + (lane & 15);
  int r0 = rb + ((lane >> 4) << 3);
  int ni = c >> 5, oo = c & 31;
  #pragma unroll
  for (int v = 0; v < 8; ++v) {
    int r = r0 + v;
    float d  = dl[r * M + c];
    float nr = (cRR[v] - cII[v]) / d;
    float nm = (cRI[v] + cIR[v]) / d;
    int mi = r >> 5, zb = r & 31;
    int P = off3(l) + mi * LL + ni;
    int idx = P * 1024 + zb * 32 + oo;
    FzR[idx] = nr; FzI[idx] = nm;
  }
}

// ---- K6: SO3 inverse: G = sum_l Fz*D; out = Re(E^T G E) + bias (WMMA) ----
__global__ void __launch_bounds__(32) k_so3(const float* __restrict__ FzR,
                                            const float* __restrict__ FzI,
                                            const float* __restrict__ Dt,
                                            const float* __restrict__ Ec,
                                            const float* __restrict__ Es,
                                            const float* __restrict__ bias,
                                            float* __restrict__ out) {
  __shared__ float sEc[1024], sEs[1024], sGr[1024], sGi[1024], sTr[1024], sTi[1024];
  int t = blockIdx.x;
  int j = t & 31, o = (t >> 5) & 31, z = t >> 10;
  int lane = threadIdx.x;
  for (int i = lane; i < 1024; i += 32) {
    sEc[i] = Ec[i]; sEs[i] = Es[i];
    sGr[i] = 0.f;   sGi[i] = 0.f;
  }
  __syncthreads();
  // build G[m,n] (31x31, zero-padded to 32x32)
  const float* Dj = Dt + j * NP;
  for (int e = lane; e < 961; e += 32) {
    int rm = e / 31, rn = e % 31;
    int m = rm - 15, n = rn - 15;
    int am = m < 0 ? -m : m, an = n < 0 ? -n : n;
    int lmin = am > an ? am : an;
    float gr = 0.f, gi = 0.f;
    for (int l = lmin; l < 16; ++l) {
      int LL = 2 * l + 1;
      int P = off3(l) + (m + l) * LL + (n + l);
      float dv = Dj[P];
      int idx = P * 1024 + z * 32 + o;
      gr = fmaf(dv, FzR[idx], gr);
      gi = fmaf(dv, FzI[idx], gi);
    }
    sGr[rm * 32 + rn] = gr;
    sGi[rm * 32 + rn] = gi;
  }
  __syncthreads();
  // Stage A: T = G * E  (transform n -> q)
  #pragma unroll
  for (int tm = 0; tm < 32; tm += 16) {
    #pragma unroll
    for (int tq = 0; tq < 32; tq += 16) {
      v8f tr = {}, ti = {};
      #pragma unroll
      for (int k0 = 0; k0 < 32; k0 += 4) {
        v2f aR = ldA(sGr, 32, tm, k0, lane);
        v2f aI = ldA(sGi, 32, tm, k0, lane);
        v2f bC = ldB(sEc, 32, tq, k0, lane);
        v2f bS = ldB(sEs, 32, tq, k0, lane);
        v2f bN; bN.x = -bS.x; bN.y = -bS.y;
        tr = wmma4(aR, bC, tr);
        tr = wmma4(aI, bN, tr);
        ti = wmma4(aR, bS, ti);
        ti = wmma4(aI, bC, ti);
      }
      int c  = tq + (lane & 15);
      int r0 = tm + ((lane >> 4) << 3);
      #pragma unroll
      for (int v = 0; v < 8; ++v) {
        sTr[(r0 + v) * 32 + c] = tr[v];
        sTi[(r0 + v) * 32 + c] = ti[v];
      }
    }
  }
  __syncthreads();
  // Stage B: out[p,q] = sum_m C[m,p]*Tr[m,q] - S[m,p]*Ti[m,q] + bias
  float bv = bias[o];
  float* ob = out + (size_t)(((z * 32 + o) * 32 + j)) * 1024;
  #pragma unroll
  for (int tp = 0; tp < 32; tp += 16) {
    #pragma unroll
    for (int tq = 0; tq < 32; tq += 16) {
      v8f acc = {bv, bv, bv, bv, bv, bv, bv, bv};
      #pragma unroll
      for (int k0 = 0; k0 < 32; k0 += 4) {
        int kk = k0 + ((lane >> 4) << 1);
        int r  = tp + (lane & 15);
        v2f aC; aC.x =  sEc[kk * 32 + r]; aC.y =  sEc[(kk + 1) * 32 + r];
        v2f aS; aS.x = -sEs[kk * 32 + r]; aS.y = -sEs[(kk + 1) * 32 + r];
        v2f bT = ldB(sTr, 32, tq, k0, lane);
        v2f bU = ldB(sTi, 32, tq, k0, lane);
        acc = wmma4(aC, bT, acc);
        acc = wmma4(aS, bU, acc);
      }
      int q  = tq + (lane & 15);
      int p0 = tp + ((lane >> 4) << 3);
      #pragma unroll
      for (int v = 0; v < 8; ++v) ob[(p0 + v) * 32 + q] = acc[v];
    }
  }
}

// ---------------- host-side launch ----------------
extern "C" void kernel_launch(void* const* d_in, const int* in_sizes, int n_in,
                              void* d_out, int out_size, void* d_ws, size_t ws_size,
                              hipStream_t stream) {
  (void)in_sizes; (void)n_in; (void)out_size; (void)ws_size;
  const float* x    = (const float*)d_in[0];   // [32,64,64,64]
  const float* kern = (const float*)d_in[1];   // [64,32,24]
  const float* bias = (const float*)d_in[2];   // [32]
  float* out = (float*)d_out;                  // [32,32,32,32,32]

  float* p   = (float*)d_ws;
  float* W   = p; p += 16384;          // W_S2 [64][256]
  float* Dt  = p; p += 32 * NP;        // D_SO3 packed [32][5456]
  float* Yre = p; p += 6144;           // Y_GRID re [24][256]
  float* Yim = p; p += 6144;
  float* twR = p; p += 1984;           // alpha-DFT twiddles [64][31]
  float* twI = p; p += 1984;
  float* Ec  = p; p += 1024;           // 32x32 Fourier table (row 31 zero)
  float* Es  = p; p += 1024;
  float* xfR = p; p += 4063232;        // xf compact [131072][31]
  float* xfI = p; p += 4063232;
  float* FxR = p; p += 524288;         // per-l Fx [32L x 64]
  float* FxI = p; p += 524288;
  float* FyR = p; p += 524288;         // per-l Fy [64 x 32L]
  float* FyI = p; p += 524288;
  float* den = p; p += 1024 * NP;      // per-l den [32L x 32L]
  float* FzR = p; p += 1024 * NP;      // packed [P][z][o]
  float* FzI = p; p += 1024 * NP;

  k_tables<<<782, 256, 0, stream>>>(W, Dt, Yre, Yim, twR, twI, Ec, Es);
  k_dft<<<15872, 256, 0, stream>>>(x, twR, twI, xfR, xfI);
  k_project<<<2048, 256, 0, stream>>>(W, xfR, xfI, FxR, FxI);
  k_y<<<2048, 256, 0, stream>>>(kern, Yre, Yim, FyR, FyI);
  dim3 g(62, 62, 16);
  k_den<<<g, 32, 0, stream>>>(FyR, FyI, den);
  k_num<<<g, 32, 0, stream>>>(FxR, FxI, FyR, FyI, den, FzR, FzI);
  k_so3<<<32768, 32, 0, stream>>>(FzR, FzI, Dt, Ec, Es, bias, out);
}